// graph_reconstructionGIN_14035953123811
// MI455X (gfx1250) — compile-verified
//
#include <hip/hip_runtime.h>

// ===========================================================================
// GIN forward for MI455X (gfx1250). Memory-bound by the 4M-edge scatter
// (~4 GB of gather+atomic traffic -> ~180us floor at 23.3 TB/s); GEMMs are
// ~5 GFLOP total, so we run them at full f32 precision on the CDNA5 matrix
// pipe with V_WMMA_F32_16X16X4_F32 (wave32, 16x16 tile per wave).
// ===========================================================================

typedef float v2f __attribute__((ext_vector_type(2)));
typedef float v8f __attribute__((ext_vector_type(8)));

#define NNODES 262144
#define NEDGES (4 * 1024 * 1024)
#define LL 4

// ---------------------------------------------------------------------------
// Zero the BN batch-stat accumulators (L x {sum[32], sumsq[32]}). Must run
// every launch: the harness does not re-poison between graph replays and we
// accumulate atomically.
__global__ void k_zero_stats(float* __restrict__ stats) {
  int i = blockIdx.x * blockDim.x + threadIdx.x;
  if (i < LL * 64) stats[i] = 0.0f;
}

// ---------------------------------------------------------------------------
// h0[i, 0:16] = [x[i,0:8], t, 0...]: pad to K=16 so layer-0 WMMA K-loop and
// edge scatter use a 16-float row (zeros aggregate to zero -> exact).
__global__ void k_build_h0(const float* __restrict__ x,
                           const float* __restrict__ t,
                           float* __restrict__ h0) {
  int idx = blockIdx.x * blockDim.x + threadIdx.x;  // N*16 threads
  int i = idx >> 4, c = idx & 15;
  float v = 0.0f;
  if (c < 8) v = x[i * 8 + c];
  else if (c == 8) v = t[0];
  h0[idx] = v;
}

// ---------------------------------------------------------------------------
// agg = (1 + eps[l]) * h_in  -- folds the GIN self term into the scatter
// accumulator so the MLP kernel reads a single buffer.
__global__ void k_init_agg(const float* __restrict__ h_in, int hstride,
                           float* __restrict__ agg, int dimshift,
                           const float* __restrict__ eps, int l) {
  int idx = blockIdx.x * blockDim.x + threadIdx.x;  // N * dim threads
  int i = idx >> dimshift;
  int c = idx & ((1 << dimshift) - 1);
  agg[idx] = (1.0f + eps[l]) * h_in[(long long)i * hstride + c];
}

// ---------------------------------------------------------------------------
// Edge scatter-add: agg[dst] += h_in[src]. One thread per (edge, 4-float
// group): float4 gather (each edge touches one 64/128B line) + 4 no-return
// global_atomic_add_f32. Prefetch a future edge's gather line to cover the
// random-access latency.
__global__ __launch_bounds__(256)
void k_scatter(const int* __restrict__ src, const int* __restrict__ dst,
               const float* __restrict__ h_in, int hstride,
               float* __restrict__ agg, int dimshift) {
  const int gshift = dimshift - 2;            // groups per edge = dim/4
  long long idx = (long long)blockIdx.x * blockDim.x + threadIdx.x;
  long long total = (long long)NEDGES << gshift;
  if (idx >= total) return;
  int e = (int)(idx >> gshift);
  int g = (int)(idx & ((1 << gshift) - 1));
  if (g == 0 && e + 256 < NEDGES) {
    __builtin_prefetch(h_in + (long long)src[e + 256] * hstride, 0, 0);
  }
  int s = src[e];
  int d = dst[e];
  const float4 v =
      *reinterpret_cast<const float4*>(h_in + (long long)s * hstride + g * 4);
  float* ap = agg + ((long long)d << dimshift) + g * 4;
  atomicAdd(ap + 0, v.x);
  atomicAdd(ap + 1, v.y);
  atomicAdd(ap + 2, v.z);
  atomicAdd(ap + 3, v.w);
}

// ---------------------------------------------------------------------------
// Fused GIN MLP: z = relu(agg @ W1 + b1) @ W2 + b2, one 16-row tile per
// wave32, 8 waves/block. Full-precision V_WMMA_F32_16X16X4_F32.
//   A-frag: a[v] = A[M = lane&15][K = 2*(lane>>4) + v]
//   B-frag: b[v] = B[K = 2*(lane>>4) + v][N = lane&15]
//   C/D   : c[r] = C[M = r + 8*(lane>>4)][N = lane&15]
// W1 is zero-padded in LDS to kin rows (layer 0: 9 real rows -> K=16).
// BN sum/sumsq fused via ds_add_f32 LDS reduction + 64 global atomics/block.
__global__ __launch_bounds__(256)
void k_gin_mlp(const float* __restrict__ agg, int kin, int k1rows,
               const float* __restrict__ W1, const float* __restrict__ b1,
               const float* __restrict__ W2, const float* __restrict__ b2,
               float* __restrict__ z, float* __restrict__ stats) {
  __shared__ float sW1[32 * 32];
  __shared__ float sW2[32 * 32];
  __shared__ float sb1[32], sb2[32];
  __shared__ float sred[64];
  __shared__ float sz1[8][16][33];  // per-wave relu(GEMM1) tile, padded stride

  int tid = threadIdx.x;
  for (int i = tid; i < kin * 32; i += 256) {
    int kk = i >> 5;
    sW1[i] = (kk < k1rows) ? W1[kk * 32 + (i & 31)] : 0.0f;
  }
  for (int i = tid; i < 1024; i += 256) sW2[i] = W2[i];
  if (tid < 32) { sb1[tid] = b1[tid]; sb2[tid] = b2[tid]; }
  if (tid < 64) sred[tid] = 0.0f;
  __syncthreads();

  int wave = tid >> 5;
  int lane = tid & 31;
  int row  = lane & 15;
  int kh   = lane >> 4;
  long long m0 = ((long long)blockIdx.x * 8 + wave) * 16;
  const float* Ain = agg + m0 * kin;

  // ---- GEMM1 + ReLU -> LDS (transpose C-layout back to row-major for GEMM2)
  for (int ct = 0; ct < 2; ++ct) {
    v8f c;
    float bv = sb1[ct * 16 + row];
#pragma unroll
    for (int r = 0; r < 8; ++r) c[r] = bv;
    for (int k0 = 0; k0 < kin; k0 += 4) {
      int ka = k0 + 2 * kh;
      v2f a, b;
      a.x = Ain[row * kin + ka];
      a.y = Ain[row * kin + ka + 1];
      b.x = sW1[ka * 32 + ct * 16 + row];
      b.y = sW1[(ka + 1) * 32 + ct * 16 + row];
      c = __builtin_amdgcn_wmma_f32_16x16x4_f32(false, a, false, b,
                                                (short)0, c, false, false);
    }
#pragma unroll
    for (int r = 0; r < 8; ++r) {
      float v = c[r] > 0.0f ? c[r] : 0.0f;
      sz1[wave][r + 8 * kh][ct * 16 + row] = v;  // intra-wave only: no barrier
    }
  }

  // ---- GEMM2 -> z (global) + fused BN batch-stat accumulation
  for (int ct = 0; ct < 2; ++ct) {
    v8f c;
    float bv = sb2[ct * 16 + row];
#pragma unroll
    for (int r = 0; r < 8; ++r) c[r] = bv;
#pragma unroll
    for (int k0 = 0; k0 < 32; k0 += 4) {
      int ka = k0 + 2 * kh;
      v2f a, b;
      a.x = sz1[wave][row][ka];
      a.y = sz1[wave][row][ka + 1];
      b.x = sW2[ka * 32 + ct * 16 + row];
      b.y = sW2[(ka + 1) * 32 + ct * 16 + row];
      c = __builtin_amdgcn_wmma_f32_16x16x4_f32(false, a, false, b,
                                                (short)0, c, false, false);
    }
    float s = 0.0f, q = 0.0f;
#pragma unroll
    for (int r = 0; r < 8; ++r) {
      float v = c[r];
      z[(m0 + r + 8 * kh) * 32 + ct * 16 + row] = v;
      s += v;
      q += v * v;
    }
    atomicAdd(&sred[ct * 16 + row], s);        // ds_add_f32
    atomicAdd(&sred[32 + ct * 16 + row], q);
  }
  __syncthreads();
  if (tid < 64) atomicAdd(&stats[tid], sred[tid]);
}

// ---------------------------------------------------------------------------
// derived = {scale = gamma * rsqrt(var + 1e-5), shift = beta - mu * scale}
__global__ void k_bn_derive(const float* __restrict__ stats,
                            const float* __restrict__ gamma,
                            const float* __restrict__ beta, int l,
                            float* __restrict__ derived) {
  int ch = threadIdx.x;  // 32
  const float invn = 1.0f / (float)NNODES;
  float mu = stats[ch] * invn;
  float var = stats[32 + ch] * invn - mu * mu;
  float sc = gamma[l * 32 + ch] * rsqrtf(var + 1e-5f);
  derived[ch] = sc;
  derived[32 + ch] = beta[l * 32 + ch] - mu * sc;
}

// ---------------------------------------------------------------------------
// h = relu(z*scale + shift) written into the JK concat buffer column slice;
// this slice is also the next layer's gather source (row stride 128).
__global__ void k_bn_apply(const float* __restrict__ z,
                           const float* __restrict__ derived,
                           float* __restrict__ xs_cat, int l) {
  int idx = blockIdx.x * blockDim.x + threadIdx.x;  // N*32 threads
  int i = idx >> 5, ch = idx & 31;
  float v = z[idx] * derived[ch] + derived[32 + ch];
  xs_cat[(long long)i * 128 + l * 32 + ch] = v > 0.0f ? v : 0.0f;
}

// ---------------------------------------------------------------------------
// Final: new_x = xs_cat @ lin_W + lin_b (K=128, 8 valid output cols padded to
// a 16-col WMMA tile) fused with the masked feature write-back.
__global__ __launch_bounds__(256)
void k_final(const float* __restrict__ xs_cat, const float* __restrict__ linW,
             const float* __restrict__ linb, const float* __restrict__ x,
             const unsigned char* __restrict__ node_mask,
             const unsigned char* __restrict__ edge_mask,
             const int* __restrict__ ond_p, const int* __restrict__ oed_p,
             float* __restrict__ out) {
  int tid = threadIdx.x;
  int wave = tid >> 5, lane = tid & 31;
  int row = lane & 15, kh = lane >> 4;
  long long m0 = ((long long)blockIdx.x * 8 + wave) * 16;
  const float* A = xs_cat + m0 * 128;

  v8f c;
  float bv = (row < 8) ? linb[row] : 0.0f;
#pragma unroll
  for (int r = 0; r < 8; ++r) c[r] = bv;
#pragma unroll 8
  for (int k0 = 0; k0 < 128; k0 += 4) {
    int ka = k0 + 2 * kh;
    v2f a, b;
    a.x = A[row * 128 + ka];
    a.y = A[row * 128 + ka + 1];
    b.x = (row < 8) ? linW[ka * 8 + row] : 0.0f;        // zero-pad cols 8..15
    b.y = (row < 8) ? linW[(ka + 1) * 8 + row] : 0.0f;
    c = __builtin_amdgcn_wmma_f32_16x16x4_f32(false, a, false, b,
                                              (short)0, c, false, false);
  }

  int ond = ond_p[0], oed = oed_p[0];
  if (row < 8) {                 // divergence only after all WMMAs retire
    int j = row;
    bool ncol = (j >= 1) && (j < ond + 1);
    bool ecol = (j >= 1) && (j < oed + 1);
#pragma unroll
    for (int r = 0; r < 8; ++r) {
      long long i = m0 + r + 8 * kh;
      float val = x[i * 8 + j];
      bool nm = node_mask[i] != 0;  // JAX bool = 1 byte
      bool em = edge_mask[i] != 0;
      if ((nm && ncol) || (em && ecol)) val = c[r];
      out[i * 8 + j] = val;
    }
  }
}

// ===========================================================================
extern "C" void kernel_launch(void* const* d_in, const int* in_sizes, int n_in,
                              void* d_out, int out_size, void* d_ws,
                              size_t ws_size, hipStream_t stream) {
  (void)in_sizes; (void)n_in; (void)out_size; (void)ws_size;
  const float* x     = (const float*)d_in[0];
  const float* t     = (const float*)d_in[1];
  const int*   eidx  = (const int*)d_in[2];
  const unsigned char* node_mask = (const unsigned char*)d_in[3];
  const unsigned char* edge_mask = (const unsigned char*)d_in[4];
  const int*   ond   = (const int*)d_in[5];
  const int*   oed   = (const int*)d_in[6];
  const float* W1f   = (const float*)d_in[7];
  const float* b1f   = (const float*)d_in[8];
  const float* W2f   = (const float*)d_in[9];
  const float* b2f   = (const float*)d_in[10];
  const float* W1r   = (const float*)d_in[11];
  const float* b1r   = (const float*)d_in[12];
  const float* W2r   = (const float*)d_in[13];
  const float* b2r   = (const float*)d_in[14];
  const float* eps   = (const float*)d_in[15];
  const float* gamma = (const float*)d_in[16];
  const float* beta  = (const float*)d_in[17];
  const float* linW  = (const float*)d_in[18];
  const float* linb  = (const float*)d_in[19];

  // Workspace layout (floats): ~218 MB total.
  float* ws      = (float*)d_ws;
  float* h0      = ws;                               // N*16
  float* agg     = h0 + (size_t)NNODES * 16;         // N*32
  float* z       = agg + (size_t)NNODES * 32;        // N*32
  float* xs      = z + (size_t)NNODES * 32;          // N*128 (JK concat)
  float* stats   = xs + (size_t)NNODES * 128;        // L*64 {sum,sumsq}
  float* derived = stats + LL * 64;                  // L*64 {scale,shift}

  const int* src = eidx;
  const int* dst = eidx + NEDGES;

  k_zero_stats<<<1, 256, 0, stream>>>(stats);
  k_build_h0<<<(NNODES * 16) / 256, 256, 0, stream>>>(x, t, h0);

  for (int l = 0; l < LL; ++l) {
    const float* h_in = (l == 0) ? h0 : (xs + (l - 1) * 32);
    int hstride  = (l == 0) ? 16 : 128;
    int dimshift = (l == 0) ? 4 : 5;
    int dim      = 1 << dimshift;

    k_init_agg<<<(NNODES * dim) / 256, 256, 0, stream>>>(h_in, hstride, agg,
                                                         dimshift, eps, l);
    long long totalS = (long long)NEDGES * (dim / 4);
    k_scatter<<<(unsigned)(totalS / 256), 256, 0, stream>>>(
        src, dst, h_in, hstride, agg, dimshift);

    const float* W1 = (l == 0) ? W1f : W1r + (size_t)(l - 1) * 1024;
    const float* B1 = (l == 0) ? b1f : b1r + (size_t)(l - 1) * 32;
    const float* W2 = (l == 0) ? W2f : W2r + (size_t)(l - 1) * 1024;
    const float* B2 = (l == 0) ? b2f : b2r + (size_t)(l - 1) * 32;
    int k1rows = (l == 0) ? 9 : 32;

    k_gin_mlp<<<NNODES / 128, 256, 0, stream>>>(agg, dim, k1rows, W1, B1, W2,
                                                B2, z, stats + l * 64);
    k_bn_derive<<<1, 32, 0, stream>>>(stats + l * 64, gamma, beta, l,
                                      derived + l * 64);
    k_bn_apply<<<(NNODES * 32) / 256, 256, 0, stream>>>(z, derived + l * 64,
                                                        xs, l);
  }

  k_final<<<NNODES / 128, 256, 0, stream>>>(xs, linW, linb, x, node_mask,
                                            edge_mask, ond, oed,
                                            (float*)d_out);
}